// JTMPN_91242285236231
// MI455X (gfx1250) — compile-verified
//
#include <hip/hip_runtime.h>

#define A_N 100000
#define B_N 200000
#define M_N 50000
#define H_N 128
#define NMOL_N 4000

typedef float v2f __attribute__((ext_vector_type(2)));
typedef float v4f __attribute__((ext_vector_type(4)));
typedef float v8f __attribute__((ext_vector_type(8)));

__device__ __forceinline__ v8f wmma_f32_k4(v2f a, v2f b, v8f c) {
    // D = A(16x4,f32) * B(4x16,f32) + C(16x16,f32), exact fp32 matrix op
    return __builtin_amdgcn_wmma_f32_16x16x4_f32(false, a, false, b, 0, c, false, false);
}

// W-chunk strides (transposed layouts: [col][k]); even -> 8B-aligned b64 frags
#define WT_S 34   // 32-row K chunks (gcd(34,64)=2: conflict-free b64 reads)
#define WI_S 42   // 40-row W_i      (gcd(42,64)=2)
#define NEI_S 132 // nei tile row stride
#define AT_S 172  // unified-A tile row stride (k_atom_out)

// ---------------------------------------------------------------------------
// Kernel 1: binput = fbonds @ W_i  ;  g0 = relu(binput)
// 64 bonds/block, 8 waves -> 4 row-tiles x 2 col-halves (4 16x16 tiles/wave)
// ---------------------------------------------------------------------------
__global__ __launch_bounds__(256) void k_bond_input(
    const float* __restrict__ fbonds, const float* __restrict__ W_i,
    float* __restrict__ binput, float* __restrict__ g0) {
    __shared__ float lds_fb[64 * 44];       // 64 x 40
    __shared__ float lds_Wt[128 * WI_S];    // W_i^T : [col][k], k=0..39
    const int tid = threadIdx.x;
    const int b0  = blockIdx.x * 64;

    for (int e = tid; e < 64 * 40; e += 256) {
        int r = e / 40, c = e % 40;
        lds_fb[r * 44 + c] = fbonds[(size_t)(b0 + r) * 40 + c];
    }
    for (int e = tid; e < 40 * 128; e += 256) {
        int r = e >> 7, c = e & 127;            // r = k-row, c = out col
        lds_Wt[c * WI_S + r] = W_i[e];
    }
    __syncthreads();

    const int lane = tid & 31, w = tid >> 5;
    const int lr = lane & 15, hi = lane >> 4;
    const int mt = w >> 1, nh = w & 1;
    const int arow = mt * 16 + lr;

    v8f acc[4] = {};
    for (int k0 = 0; k0 < 40; k0 += 4) {
        int ka = k0 + 2 * hi;
        v2f a; a.x = lds_fb[arow * 44 + ka]; a.y = lds_fb[arow * 44 + ka + 1];
#pragma unroll
        for (int ct = 0; ct < 4; ++ct) {
            int col = nh * 64 + ct * 16 + lr;
            v2f b = *(const v2f*)&lds_Wt[col * WI_S + ka];   // single b64 load
            acc[ct] = wmma_f32_k4(a, b, acc[ct]);
        }
    }
#pragma unroll
    for (int ct = 0; ct < 4; ++ct) {
        int col = nh * 64 + ct * 16 + lr;
#pragma unroll
        for (int r = 0; r < 8; ++r) {
            size_t off = (size_t)(b0 + mt * 16 + hi * 8 + r) * H_N + col;
            float v = acc[ct][r];
            binput[off] = v;
            g0[off] = v > 0.f ? v : 0.f;
        }
    }
}

// ---------------------------------------------------------------------------
// Kernel 2: one message-passing step (fused gather + GEMM + bias + relu)
// gout = relu(binput + (sum_k msg[bgraph[b][k]]) @ W_h)
// ---------------------------------------------------------------------------
__global__ __launch_bounds__(256) void k_mp(
    const float* __restrict__ tree, const float* __restrict__ gin,
    const float* __restrict__ binput, const float* __restrict__ W_h,
    const int* __restrict__ bgraph, float* __restrict__ gout) {
    __shared__ float lds_nei[64 * NEI_S];   // 64 x 128
    __shared__ float lds_Wt[128 * WT_S];    // transposed 32-row K chunk of W_h
    __shared__ int   s_idx[64 * 8];
    const int tid = threadIdx.x;
    const int b0  = blockIdx.x * 64;

    for (int e = tid; e < 512; e += 256) s_idx[e] = bgraph[(size_t)b0 * 8 + e];
    __syncthreads();

    // gather + neighbor sum, float4-wide (L2-resident message table)
    for (int e = tid; e < 64 * 32; e += 256) {
        int r = e >> 5, c4 = (e & 31) * 4;
        v4f s = {};
#pragma unroll
        for (int k = 0; k < 8; ++k) {
            int idx = s_idx[r * 8 + k];
            const float* src = (idx < M_N) ? (tree + (size_t)idx * H_N)
                                           : (gin + (size_t)(idx - M_N) * H_N);
            s += *(const v4f*)(src + c4);
        }
        *(v4f*)&lds_nei[r * NEI_S + c4] = s;
    }

    const int lane = tid & 31, w = tid >> 5;
    const int lr = lane & 15, hi = lane >> 4;
    const int mt = w >> 1, nh = w & 1;
    const int arow = mt * 16 + lr;

    v8f acc[4] = {};
    for (int kc = 0; kc < 128; kc += 32) {
        __syncthreads();   // (also orders nei fill before first use)
        for (int e = tid; e < 32 * 128; e += 256) {
            int r = e >> 7, c = e & 127;
            lds_Wt[c * WT_S + r] = W_h[(size_t)(kc + r) * H_N + c];
        }
        __syncthreads();
        for (int k0 = 0; k0 < 32; k0 += 4) {
            int ka = k0 + 2 * hi;
            v2f a = *(const v2f*)&lds_nei[arow * NEI_S + kc + ka];
#pragma unroll
            for (int ct = 0; ct < 4; ++ct) {
                int col = nh * 64 + ct * 16 + lr;
                v2f b = *(const v2f*)&lds_Wt[col * WT_S + ka];
                acc[ct] = wmma_f32_k4(a, b, acc[ct]);
            }
        }
    }
#pragma unroll
    for (int ct = 0; ct < 4; ++ct) {
        int col = nh * 64 + ct * 16 + lr;
#pragma unroll
        for (int r = 0; r < 8; ++r) {
            size_t off = (size_t)(b0 + mt * 16 + hi * 8 + r) * H_N + col;
            float v = binput[off] + acc[ct][r];
            gout[off] = v > 0.f ? v : 0.f;
        }
    }
}

// ---------------------------------------------------------------------------
// Kernel 3: atom readout. Unified K permuted as:
//   k 0..127  -> gathered nei  (x W_o rows 35..162)
//   k 128..162-> fatoms        (x W_o rows 0..34)
//   k 163     -> 1.0 bias col  (x b_o)
// atom_hiddens = relu(A @ W) ; fp32 atomic segment sums + counts
// ---------------------------------------------------------------------------
__global__ __launch_bounds__(256) void k_atom_out(
    const float* __restrict__ fatoms, const float* __restrict__ tree,
    const float* __restrict__ gfin, const float* __restrict__ W_o,
    const float* __restrict__ b_o, const int* __restrict__ agraph,
    const int* __restrict__ mol_id,
    float* __restrict__ sums, float* __restrict__ counts) {
    __shared__ float lds_A[64 * AT_S];      // 64 x 164 (permuted K)
    __shared__ float lds_Wt[128 * WT_S];    // transposed 32-row K chunk
    __shared__ int   s_idx[64 * 8];
    __shared__ int   s_mol[64];
    const int tid = threadIdx.x;
    const int a0  = blockIdx.x * 64;

    for (int e = tid; e < 512; e += 256) {
        int atom = a0 + (e >> 3);
        s_idx[e] = (atom < A_N) ? agraph[(size_t)a0 * 8 + e] : 0;  // idx 0 = zero msg
    }
    if (tid < 64) {
        int atom = a0 + tid;
        s_mol[tid] = (atom < A_N) ? mol_id[atom] : 0;
        lds_A[tid * AT_S + 163] = 1.0f;     // bias column
    }
    for (int e = tid; e < 64 * 35; e += 256) {
        int r = e / 35, c = e % 35;
        int atom = a0 + r;
        lds_A[r * AT_S + 128 + c] = (atom < A_N) ? fatoms[(size_t)atom * 35 + c] : 0.f;
    }
    __syncthreads();
    for (int e = tid; e < 64 * 32; e += 256) {
        int r = e >> 5, c4 = (e & 31) * 4;
        v4f s = {};
#pragma unroll
        for (int k = 0; k < 8; ++k) {
            int idx = s_idx[r * 8 + k];
            const float* src = (idx < M_N) ? (tree + (size_t)idx * H_N)
                                           : (gfin + (size_t)(idx - M_N) * H_N);
            s += *(const v4f*)(src + c4);
        }
        *(v4f*)&lds_A[r * AT_S + c4] = s;
    }

    const int lane = tid & 31, w = tid >> 5;
    const int lr = lane & 15, hi = lane >> 4;
    const int mt = w >> 1, nh = w & 1;
    const int arow = mt * 16 + lr;

    v8f acc[4] = {};
    for (int kc = 0; kc < 164; kc += 32) {
        const int klen = (164 - kc) < 32 ? (164 - kc) : 32;
        __syncthreads();
        for (int e = tid; e < 32 * 128; e += 256) {
            int r = e >> 7, c = e & 127;
            int gk = kc + r;                 // permuted K row -> W_o source row
            float wv;
            if (gk < 128)      wv = W_o[(size_t)(35 + gk) * H_N + c];
            else if (gk < 163) wv = W_o[(size_t)(gk - 128) * H_N + c];
            else if (gk == 163) wv = b_o[c];
            else               wv = 0.f;
            lds_Wt[c * WT_S + r] = wv;
        }
        __syncthreads();
        for (int k0 = 0; k0 < klen; k0 += 4) {
            int ka = k0 + 2 * hi;
            v2f a = *(const v2f*)&lds_A[arow * AT_S + kc + ka];
#pragma unroll
            for (int ct = 0; ct < 4; ++ct) {
                int col = nh * 64 + ct * 16 + lr;
                v2f b = *(const v2f*)&lds_Wt[col * WT_S + ka];
                acc[ct] = wmma_f32_k4(a, b, acc[ct]);
            }
        }
    }
#pragma unroll
    for (int ct = 0; ct < 4; ++ct) {
        int col = nh * 64 + ct * 16 + lr;
#pragma unroll
        for (int r = 0; r < 8; ++r) {
            int lrow = mt * 16 + hi * 8 + r;
            if (a0 + lrow < A_N) {
                float v = acc[ct][r];
                v = v > 0.f ? v : 0.f;
                atomicAdd(&sums[(size_t)s_mol[lrow] * H_N + col], v);
            }
        }
    }
    if (nh == 0 && lr == 0) {   // lanes 0 & 16 of waves 0,2,4,6 -> counts once/atom
#pragma unroll
        for (int r = 0; r < 8; ++r) {
            int lrow = mt * 16 + hi * 8 + r;
            if (a0 + lrow < A_N) atomicAdd(&counts[s_mol[lrow]], 1.0f);
        }
    }
}

__global__ __launch_bounds__(256) void k_zero(float* __restrict__ p, int n) {
    int i = blockIdx.x * 256 + threadIdx.x;
    if (i < n) p[i] = 0.f;
}

__global__ __launch_bounds__(256) void k_final(const float* __restrict__ sums,
                                               const float* __restrict__ counts,
                                               float* __restrict__ out) {
    int i = blockIdx.x * 256 + threadIdx.x;
    if (i < NMOL_N * H_N) out[i] = sums[i] / fmaxf(counts[i >> 7], 1.0f);
}

extern "C" void kernel_launch(void* const* d_in, const int* in_sizes, int n_in,
                              void* d_out, int out_size, void* d_ws, size_t ws_size,
                              hipStream_t stream) {
    (void)in_sizes; (void)n_in; (void)out_size; (void)ws_size;
    const float* fatoms = (const float*)d_in[0];
    const float* fbonds = (const float*)d_in[1];
    const float* tree   = (const float*)d_in[2];
    const int*   agraph = (const int*)d_in[3];
    const int*   bgraph = (const int*)d_in[4];
    const int*   mol_id = (const int*)d_in[5];
    const float* W_i    = (const float*)d_in[6];
    const float* W_h    = (const float*)d_in[7];
    const float* W_o    = (const float*)d_in[8];
    const float* b_o    = (const float*)d_in[9];

    float* binput = (float*)d_ws;                       // [B, H]
    float* g0     = binput + (size_t)B_N * H_N;         // [B, H]
    float* g1     = g0 + (size_t)B_N * H_N;             // [B, H]
    float* sums   = g1 + (size_t)B_N * H_N;             // [NMOL, H]
    float* counts = sums + (size_t)NMOL_N * H_N;        // [NMOL]

    const int nz = NMOL_N * H_N + NMOL_N;               // sums+counts contiguous
    k_zero<<<(nz + 255) / 256, 256, 0, stream>>>(sums, nz);

    k_bond_input<<<B_N / 64, 256, 0, stream>>>(fbonds, W_i, binput, g0);

    // DEPTH-1 = 4 message-passing iterations, double-buffered
    k_mp<<<B_N / 64, 256, 0, stream>>>(tree, g0, binput, W_h, bgraph, g1);
    k_mp<<<B_N / 64, 256, 0, stream>>>(tree, g1, binput, W_h, bgraph, g0);
    k_mp<<<B_N / 64, 256, 0, stream>>>(tree, g0, binput, W_h, bgraph, g1);
    k_mp<<<B_N / 64, 256, 0, stream>>>(tree, g1, binput, W_h, bgraph, g0);

    k_atom_out<<<(A_N + 63) / 64, 256, 0, stream>>>(fatoms, tree, g0, W_o, b_o,
                                                    agraph, mol_id, sums, counts);

    k_final<<<(NMOL_N * H_N + 255) / 256, 256, 0, stream>>>(sums, counts, (float*)d_out);
}